// MLPWithClustering_23330262352470
// MI455X (gfx1250) — compile-verified
//
#include <hip/hip_runtime.h>
#include <math.h>
#include <stdint.h>

#define BROWS 16384
#define DDIM  1024
#define NCLUST 15
#define NPAD 16
#define CCLS 1000
#define KTH_RANK 6554          // 16384 - int(16384*0.6)
#define ALPHA_C 5.0f
#define LAM_CLUSTER 6.0f
#define LAM_DIV 0.1f
#define EPS_NORM 1e-12f
#define EPS_LOG 1e-8f

typedef __attribute__((ext_vector_type(2))) float v2f;
typedef __attribute__((ext_vector_type(8))) float v8f;

// ---------------- workspace layout (floats) ----------------
// centT  : [3][1024][16]   offset 0        (49152)
// gsum   : [3][15][1000]   offset 49152    (45000)
// gcnt   : [3][15]         offset 94152    (45)   -- pad to 94208
// avg    : [3][16384]      offset 94208    (49152)
// assign : [3][16384] int  offset 143360   (49152)

__global__ void zero_kernel(float* p, int n) {
    int i = blockIdx.x * blockDim.x + threadIdx.x;
    if (i < n) p[i] = 0.0f;
}

__global__ void copy_kernel(const float4* __restrict__ src, float4* __restrict__ dst, int n) {
    int i = blockIdx.x * blockDim.x + threadIdx.x;
    if (i < n) dst[i] = src[i];
}

// Normalize centroid rows; store transposed [D][16] per modality, col 15 left zero.
__global__ __launch_bounds__(256) void centnorm_kernel(const float* __restrict__ c0,
                                                       const float* __restrict__ c1,
                                                       const float* __restrict__ c2,
                                                       float* __restrict__ centT) {
    const int m = blockIdx.x / NCLUST;
    const int k = blockIdx.x % NCLUST;
    const float* c = (m == 0) ? c0 : ((m == 1) ? c1 : c2);
    __shared__ float red[256];
    const int tid = threadIdx.x;
    float s = 0.0f;
    for (int d = tid; d < DDIM; d += 256) { float v = c[k * DDIM + d]; s += v * v; }
    red[tid] = s; __syncthreads();
    for (int o = 128; o > 0; o >>= 1) { if (tid < o) red[tid] += red[tid + o]; __syncthreads(); }
    const float inv = 1.0f / fmaxf(sqrtf(red[0]), EPS_NORM);
    float* dst = centT + (size_t)m * DDIM * NPAD;
    for (int d = tid; d < DDIM; d += 256) dst[d * NPAD + k] = c[k * DDIM + d] * inv;
}

// 64 rows per block, 4 waves (wave32), each wave owns 16 rows.
// fea is streamed HBM -> LDS with double-buffered GLOBAL_LOAD_ASYNC_TO_LDS_B128
// (ASYNCcnt), overlapping the next chunk's DMA with the current chunk's
// V_WMMA_F32_16X16X4_F32 work. Fused row-norm / mean / argmax.
__global__ __launch_bounds__(128) void sim_kernel(const float* __restrict__ f0,
                                                  const float* __restrict__ f1,
                                                  const float* __restrict__ f2,
                                                  const float* __restrict__ centT,
                                                  float* __restrict__ avg_sim,
                                                  int* __restrict__ assign) {
    const int m = blockIdx.y;
    const float* fea  = (m == 0) ? f0 : ((m == 1) ? f1 : f2);
    const float* bmat = centT + (size_t)m * DDIM * NPAD;
    const int rowbase = blockIdx.x * 64;

    __shared__ float tile[2][64][36];  // double-buffered 64x32-K chunk (pad 36: 144B stride)
    __shared__ float sumsq[64];
    __shared__ float simbuf[64][17];

    const int tid   = threadIdx.x;
    const int wave  = tid >> 5;
    const int lane  = tid & 31;
    const int wrow  = wave * 16;
    const int lhalf = lane >> 4;      // 0 or 1
    const int l15   = lane & 15;

    const int srow = tid >> 3;        // staging base row (0..15)
    const int sc4  = tid & 7;         // float4 column within 32-float chunk

    if (tid < 64) sumsq[tid] = 0.0f;

    // LDS byte offset of tile (flat-LDS translation uses addr[31:0])
    const uint32_t ldsbase = (uint32_t)(uintptr_t)(&tile[0][0][0]);
    const float* gbase = fea + (size_t)rowbase * DDIM;

    // prologue: async-stage chunk 0 into buffer 0
    #pragma unroll
    for (int i = 0; i < 4; ++i) {
        const int row = srow + 16 * i;
        const uint64_t ga = (uint64_t)(uintptr_t)(gbase + (size_t)row * DDIM + sc4 * 4);
        const uint32_t la = ldsbase + (uint32_t)((row * 36 + sc4 * 4) * 4);
        asm volatile("global_load_async_to_lds_b128 %0, %1, off"
                     :: "v"(la), "v"(ga) : "memory");
    }

    v8f acc = {};
    float ssq[4] = {0.f, 0.f, 0.f, 0.f};
    int cur = 0;

    for (int kc = 0; kc < DDIM; kc += 32) {
        const int nxt = kc + 32;
        if (nxt < DDIM) {
            // issue async DMA for the next chunk into the other buffer
            const int nbuf = cur ^ 1;
            #pragma unroll
            for (int i = 0; i < 4; ++i) {
                const int row = srow + 16 * i;
                const uint64_t ga = (uint64_t)(uintptr_t)(gbase + (size_t)row * DDIM + nxt + sc4 * 4);
                const uint32_t la = ldsbase + (uint32_t)(((nbuf * 64 + row) * 36 + sc4 * 4) * 4);
                asm volatile("global_load_async_to_lds_b128 %0, %1, off"
                             :: "v"(la), "v"(ga) : "memory");
            }
            // wave issued 4 newer asyncs; loads retire in order, so <=4 outstanding
            // means the current chunk's transfers have landed in LDS.
            asm volatile("s_wait_asynccnt 0x4" ::: "memory");
        } else {
            asm volatile("s_wait_asynccnt 0x0" ::: "memory");
        }
        __syncthreads();   // cross-wave visibility of staged chunk

        // row sum-of-squares from LDS (each thread owns rows srow+16i)
        #pragma unroll
        for (int i = 0; i < 4; ++i) {
            const int row = srow + 16 * i;
            const float4 v = *(const float4*)&tile[cur][row][sc4 * 4];
            ssq[i] += v.x * v.x + v.y * v.y + v.z * v.z + v.w * v.w;
        }

        // 8 WMMA steps; A fragments from LDS, B from (L2-resident) transposed centroids.
        // A 16x4 layout: lanes 0-15 hold K={kk,kk+1}, lanes 16-31 K={kk+2,kk+3}.
        #pragma unroll
        for (int kk = 0; kk < 32; kk += 4) {
            v2f a, b;
            const float* ap = &tile[cur][wrow + l15][kk + lhalf * 2];
            a.x = ap[0];
            a.y = ap[1];
            b.x = bmat[(size_t)(kc + kk + lhalf) * NPAD + l15];
            b.y = bmat[(size_t)(kc + kk + 2 + lhalf) * NPAD + l15];
            acc = __builtin_amdgcn_wmma_f32_16x16x4_f32(
                false, a, false, b, (short)0, acc, false, false);
        }
        __syncthreads();   // protect buffer cur before it is re-filled next+1 iteration
        cur ^= 1;
    }

    atomicAdd(&sumsq[srow +  0], ssq[0]);
    atomicAdd(&sumsq[srow + 16], ssq[1]);
    atomicAdd(&sumsq[srow + 32], ssq[2]);
    atomicAdd(&sumsq[srow + 48], ssq[3]);
    __syncthreads();

    // C/D layout: VGPR r holds M=r (lanes 0-15) / M=r+8 (lanes 16-31), N=lane&15
    #pragma unroll
    for (int r = 0; r < 8; ++r) {
        const int row = wrow + r + lhalf * 8;
        const float inv = 1.0f / fmaxf(sqrtf(sumsq[row]), EPS_NORM);
        simbuf[row][l15] = acc[r] * inv;   // centroids pre-normalized -> cosine sim
    }
    __syncthreads();

    if (tid < 64) {
        const int b = rowbase + tid;
        float best = simbuf[tid][0];
        int   bi   = 0;
        float s    = best;
        for (int n = 1; n < NCLUST; ++n) {
            const float v = simbuf[tid][n];
            s += v;
            if (v > best) { best = v; bi = n; }   // strict > keeps first occurrence (argmax)
        }
        avg_sim[m * BROWS + b] = s * (1.0f / (float)NCLUST);
        assign [m * BROWS + b] = bi;
    }
}

// Fused softmax + per-cluster segment sums for all 3 modalities.
// 64 rows/block -> 256 blocks; per-block partials in 185KB dynamic LDS
// (CDNA5 WGP has 320KB). Thread t always owns columns c === t (mod 256),
// so LDS accumulation is race-free without atomics.
__global__ __launch_bounds__(256) void div_kernel(const float* __restrict__ out,
                                                  const int* __restrict__ assign,
                                                  float* __restrict__ gsum,
                                                  float* __restrict__ gcnt) {
    extern __shared__ float smem[];
    float* accum  = smem;                 // 3*15*1000 = 45000
    float* rowbuf = accum + 45000;        // 1000
    float* red    = rowbuf + CCLS;        // 256
    float* cnt    = red + 256;            // 45
    const int tid = threadIdx.x;

    for (int i = tid; i < 45000; i += 256) accum[i] = 0.0f;
    for (int i = tid; i < 45;    i += 256) cnt[i] = 0.0f;
    __syncthreads();

    const int rowbase = blockIdx.x * 64;
    for (int r = 0; r < 64; ++r) {
        const int b = rowbase + r;
        const float* orow = out + (size_t)b * CCLS;
        for (int c = tid; c < CCLS; c += 256) rowbuf[c] = orow[c];
        if (r + 1 < 64 && tid < 63) {
            __builtin_prefetch(orow + CCLS + tid * 16, 0, 0);  // next row -> global_prefetch_b8
        }
        __syncthreads();

        float mx = -3.4e38f;
        for (int c = tid; c < CCLS; c += 256) mx = fmaxf(mx, rowbuf[c]);
        red[tid] = mx; __syncthreads();
        for (int o = 128; o > 0; o >>= 1) { if (tid < o) red[tid] = fmaxf(red[tid], red[tid + o]); __syncthreads(); }
        mx = red[0]; __syncthreads();

        float s = 0.0f;
        for (int c = tid; c < CCLS; c += 256) s += expf(rowbuf[c] - mx);
        red[tid] = s; __syncthreads();
        for (int o = 128; o > 0; o >>= 1) { if (tid < o) red[tid] += red[tid + o]; __syncthreads(); }
        const float inv = 1.0f / red[0];

        const int a0 = assign[b];
        const int a1 = assign[BROWS + b];
        const int a2 = assign[2 * BROWS + b];
        for (int c = tid; c < CCLS; c += 256) {
            const float p = expf(rowbuf[c] - mx) * inv;
            accum[a0 * CCLS + c]                 += p;
            accum[(NCLUST + a1) * CCLS + c]      += p;
            accum[(2 * NCLUST + a2) * CCLS + c]  += p;
        }
        if (tid < 3) {
            const int a = (tid == 0) ? a0 : ((tid == 1) ? a1 : a2);
            cnt[tid * NCLUST + a] += 1.0f;
        }
        __syncthreads();
    }

    for (int i = tid; i < 45000; i += 256) atomicAdd(&gsum[i], accum[i]);
    for (int i = tid; i < 45;    i += 256) atomicAdd(&gcnt[i], cnt[i]);
}

// Single block: exact rank-6554 threshold via 32-pass bitwise radix-select,
// sigmoid-weighted attractive mean, per-cluster entropy, final scalar loss.
__global__ __launch_bounds__(1024) void fin_kernel(const float* __restrict__ avg_sim,
                                                   const float* __restrict__ gsum,
                                                   const float* __restrict__ gcnt,
                                                   float* __restrict__ loss_out) {
    __shared__ float red[1024];
    __shared__ unsigned ucnt;
    const int tid = threadIdx.x;

    float total_attr = 0.0f;
    for (int m = 0; m < 3; ++m) {
        const float* a = avg_sim + m * BROWS;
        unsigned fixedv = 0u, fixmask = 0u;
        int rank = KTH_RANK;
        for (int bit = 31; bit >= 0; --bit) {
            const unsigned cand = fixmask | (1u << bit);
            if (tid == 0) ucnt = 0u;
            __syncthreads();
            unsigned local = 0;
            for (int i = tid; i < BROWS; i += 1024) {
                const unsigned u = __float_as_uint(a[i]);
                const unsigned key = (u & 0x80000000u) ? ~u : (u | 0x80000000u);
                if ((key & cand) == fixedv) local++;
            }
            atomicAdd(&ucnt, local);
            __syncthreads();
            const int count = (int)ucnt;     // # candidates with this bit == 0
            __syncthreads();
            if (rank < count) { fixmask = cand; }
            else              { fixedv |= (1u << bit); fixmask = cand; rank -= count; }
        }
        const unsigned ku = (fixedv & 0x80000000u) ? (fixedv & 0x7fffffffu) : ~fixedv;
        const float thr = __uint_as_float(ku);

        float s = 0.0f;
        for (int i = tid; i < BROWS; i += 1024) {
            const float v = a[i];
            const float w = 1.0f / (1.0f + expf(-ALPHA_C * (v - thr)));
            s += w * (1.0f - v);
        }
        red[tid] = s; __syncthreads();
        for (int o = 512; o > 0; o >>= 1) { if (tid < o) red[tid] += red[tid + o]; __syncthreads(); }
        total_attr += red[0] * (1.0f / (float)BROWS);
        __syncthreads();
    }

    float total_div = 0.0f;
    for (int cl = 0; cl < 3 * NCLUST; ++cl) {
        const float cv = gcnt[cl];
        const float denom = fmaxf(cv, 1.0f);
        float s = 0.0f;
        for (int c = tid; c < CCLS; c += 1024) {
            const float p = gsum[cl * CCLS + c] / denom;
            s += p * logf(p + EPS_LOG);
        }
        red[tid] = s; __syncthreads();
        for (int o = 512; o > 0; o >>= 1) { if (tid < o) red[tid] += red[tid + o]; __syncthreads(); }
        if (cv >= 3.0f) total_div += red[0];
        __syncthreads();
    }

    if (tid == 0) loss_out[0] = LAM_CLUSTER * total_attr + LAM_DIV * total_div;
}

extern "C" void kernel_launch(void* const* d_in, const int* in_sizes, int n_in,
                              void* d_out, int out_size, void* d_ws, size_t ws_size,
                              hipStream_t stream) {
    const float* f0  = (const float*)d_in[0];   // fea_image  [B,D]
    const float* f1  = (const float*)d_in[1];   // fea_ocr
    const float* f2  = (const float*)d_in[2];   // fea_trans
    const float* c0  = (const float*)d_in[3];   // cent_image [K,D]
    const float* c1  = (const float*)d_in[4];   // cent_ocr
    const float* c2  = (const float*)d_in[5];   // cent_trans
    const float* oin = (const float*)d_in[6];   // out [B,C]

    float* ws     = (float*)d_ws;
    float* centT  = ws;                      // 49152
    float* gsum   = ws + 49152;              // 45000
    float* gcnt   = ws + 94152;              // 45
    float* avg    = ws + 94208;              // 49152
    int*   assign = (int*)(ws + 143360);     // 49152 ints

    float* outw = (float*)d_out;

    // 1) zero centT + gsum + gcnt
    zero_kernel<<<(94208 + 255) / 256, 256, 0, stream>>>(ws, 94208);

    // 2) normalized, transposed centroids (B-operand layout), col 15 stays 0
    centnorm_kernel<<<3 * NCLUST, 256, 0, stream>>>(c0, c1, c2, centT);

    // 3) WMMA skinny GEMM + async-LDS staging + fused norm/mean/argmax
    dim3 gsim(BROWS / 64, 3);
    sim_kernel<<<gsim, 128, 0, stream>>>(f0, f1, f2, centT, avg, assign);

    // 4) passthrough copy of `out` (float4, 16,384,000 % 4 == 0)
    copy_kernel<<<(BROWS * CCLS / 4 + 255) / 256, 256, 0, stream>>>(
        (const float4*)oin, (float4*)outw, BROWS * CCLS / 4);

    // 5) fused softmax + per-cluster segment sums (185KB dynamic LDS/block)
    const size_t dsm = (size_t)(45000 + CCLS + 256 + 45) * sizeof(float);
    div_kernel<<<BROWS / 64, 256, dsm, stream>>>(oin, assign, gsum, gcnt);

    // 6) scalar loss epilogue
    fin_kernel<<<1, 1024, 0, stream>>>(avg, gsum, gcnt, outw + (size_t)BROWS * CCLS);
}